// CrossEntropySoftECE_30863634989803
// MI455X (gfx1250) — compile-verified
//
#include <hip/hip_runtime.h>
#include <hip/hip_bf16.h>
#include <math.h>

typedef __attribute__((ext_vector_type(2))) float v2f;
typedef __attribute__((ext_vector_type(4))) float fv4;
typedef __attribute__((ext_vector_type(8))) float v8f;

#define NCOLS      1000
#define NBINS      15
#define SOFT_TEMP  1.1f
#define WAVES_PER_BLOCK 8
#define PART_STRIDE 52   // 0..15 sum_coeff, 16..31 conf, 32..47 acc, 48 ce-sum

// softmax over 15 anchors for prediction p, return coefficient of bin n
__device__ __forceinline__ float softbin_coeff(float p, int n) {
    const float invT = 1.0f / SOFT_TEMP;
    float d[NBINS];
    float m = -3.4e38f;
#pragma unroll
    for (int k = 0; k < NBINS; ++k) {
        float a = (float)(2 * k + 1) * (1.0f / 30.0f);
        float t = p - a;
        d[k] = -(t * t) * invT;
        m = fmaxf(m, d[k]);
    }
    float s = 0.0f;
#pragma unroll
    for (int k = 0; k < NBINS; ++k) s += __expf(d[k] - m);
    return __expf(d[n] - m) / s;
}

__global__ void ce_softece_main(const float* __restrict__ x,
                                const int* __restrict__ tgt,
                                float* __restrict__ partials,
                                int nRows) {
    __shared__ float lds[WAVES_PER_BLOCK * PART_STRIDE];

    const int tid  = threadIdx.x;
    const int lane = tid & 31;
    const int wave = tid >> 5;
    const int gw   = blockIdx.x * WAVES_PER_BLOCK + wave;
    const int nw   = gridDim.x * WAVES_PER_BLOCK;
    const int half = lane >> 4;   // K-pair selector for A/B operands
    const int nsub = lane & 15;   // bin index (B/D column) / matrix row (A)

    v8f acc = {0.f, 0.f, 0.f, 0.f, 0.f, 0.f, 0.f, 0.f};
    float ceAcc = 0.0f;

    for (int base = gw * 4; base < nRows; base += nw * 4) {
        float preds[4], labels[4];
        bool  valid[4];

#pragma unroll 1
        for (int i = 0; i < 4; ++i) {
            int row = base + i;
            valid[i]  = (row < nRows);
            preds[i]  = 0.0f;
            labels[i] = 0.0f;
            if (!valid[i]) continue;

            const float* rp = x + (long long)row * NCOLS;

            // one streaming pass: 8 x float4 per lane covers 1024 slots (1000 valid)
            fv4 v[8];
#pragma unroll
            for (int c = 0; c < 8; ++c) {
                int idx0 = c * 128 + lane * 4;
                if (idx0 + 3 < NCOLS) {
                    v[c] = __builtin_nontemporal_load((const fv4*)(rp + idx0));
                } else {
                    v[c] = (fv4){-3.4e38f, -3.4e38f, -3.4e38f, -3.4e38f};
                }
            }

            // lane-local max + first-occurrence argmax
            float m = -3.4e38f;
            int   am = 0;
#pragma unroll
            for (int c = 0; c < 8; ++c) {
                int idx0 = c * 128 + lane * 4;
#pragma unroll
                for (int j = 0; j < 4; ++j) {
                    float val = v[c][j];
                    if (val > m) { m = val; am = idx0 + j; }
                }
            }
            // wave32 butterfly: global max, ties -> smallest index (argmax-first)
#pragma unroll
            for (int off = 16; off > 0; off >>= 1) {
                float om = __shfl_xor(m, off, 32);
                int   oa = __shfl_xor(am, off, 32);
                if (om > m || (om == m && oa < am)) { m = om; am = oa; }
            }

            // sum of exp(x - max) from registers (padded entries underflow to 0)
            float ssum = 0.0f;
#pragma unroll
            for (int c = 0; c < 8; ++c) {
#pragma unroll
                for (int j = 0; j < 4; ++j) {
                    ssum += __expf(v[c][j] - m);
                }
            }
#pragma unroll
            for (int off = 16; off > 0; off >>= 1) {
                ssum += __shfl_xor(ssum, off, 32);
            }

            float lse = m + __logf(ssum);
            int   t   = tgt[row];
            float xt  = rp[t];          // single broadcast load, hits cache
            ceAcc += (lse - xt);        // identical on all lanes

            preds[i]  = (float)am;
            labels[i] = (float)t;
        }

        // --- build WMMA operands for this group of 4 samples ---
        // B (4x16): B[k, n] = coeff(sample k, bin n); vgpr0 -> K=2*half, vgpr1 -> K=2*half+1
        float b0 = 0.0f, b1 = 0.0f;
        {
            int s0 = 2 * half, s1 = s0 + 1;
            if (nsub < NBINS) {
                if (valid[s0]) b0 = softbin_coeff(preds[s0], nsub);
                if (valid[s1]) b1 = softbin_coeff(preds[s1], nsub);
            }
        }
        // A (16x4): row0 = ones, row1 = preds, row2 = labels, rest 0
        int k0 = 2 * half;
        float a0 = (nsub == 0) ? 1.0f : (nsub == 1) ? preds[k0]     : (nsub == 2) ? labels[k0]     : 0.0f;
        float a1 = (nsub == 0) ? 1.0f : (nsub == 1) ? preds[k0 + 1] : (nsub == 2) ? labels[k0 + 1] : 0.0f;

        v2f A = {a0, a1};
        v2f B = {b0, b1};
        // D[0,n]=sum coeff, D[1,n]=sum pred*coeff, D[2,n]=sum label*coeff, accumulated in C
        acc = __builtin_amdgcn_wmma_f32_16x16x4_f32(
            /*neg_a=*/false, A, /*neg_b=*/false, B,
            /*c_mod=*/(short)0, acc, /*reuse_a=*/false, /*reuse_b=*/false);
    }

    // --- per-wave partials -> LDS (lanes 0..15 hold M = vgpr index, N = lane) ---
    if (lane < 16) {
        lds[wave * PART_STRIDE + nsub]        = acc[0];  // sum_coeffs[bin]
        lds[wave * PART_STRIDE + 16 + nsub]   = acc[1];  // sum pred*coeff
        lds[wave * PART_STRIDE + 32 + nsub]   = acc[2];  // sum label*coeff
    }
    if (lane == 0) {
        lds[wave * PART_STRIDE + 48] = ceAcc;
    }
    __syncthreads();

    // --- block partials -> global (deterministic, no atomics) ---
    if (tid < 49) {
        float s = 0.0f;
#pragma unroll
        for (int w = 0; w < WAVES_PER_BLOCK; ++w) s += lds[w * PART_STRIDE + tid];
        partials[blockIdx.x * PART_STRIDE + tid] = s;
    }
}

__global__ void ce_softece_final(const float* __restrict__ partials,
                                 float* __restrict__ out,
                                 int nBlocks, int nRows) {
    __shared__ float s[49];
    int t = threadIdx.x;
    if (t < 49) {
        float a = 0.0f;
        for (int b = 0; b < nBlocks; ++b) a += partials[b * PART_STRIDE + t];
        s[t] = a;
    }
    __syncthreads();
    if (t == 0) {
        float num = 0.0f, sabs = 0.0f;
        for (int k = 0; k < NBINS; ++k) {
            float sc    = s[k];
            float denom = fmaxf(sc, 1e-5f);
            float conf  = s[16 + k] / denom;
            float accb  = s[32 + k] / denom;
            float diff  = conf - accb;
            num  += diff * diff * sc;   // (conf-acc)^2 * sum_coeff
            sabs += fabsf(sc);          // l1 norm of bin mass
        }
        float ece = sqrtf(num / sabs);
        float ce  = s[48] / (float)nRows;
        out[0] = ce + 0.5f * ece;
    }
}

extern "C" void kernel_launch(void* const* d_in, const int* in_sizes, int n_in,
                              void* d_out, int out_size, void* d_ws, size_t ws_size,
                              hipStream_t stream) {
    const float* x   = (const float*)d_in[0];
    const int*   tgt = (const int*)d_in[1];
    float*       out = (float*)d_out;
    int nRows = in_sizes[1];

    float* partials = (float*)d_ws;
    int maxBlocks = (int)(ws_size / (PART_STRIDE * sizeof(float)));
    int nBlocks = 512;                    // 4096 waves; each wave streams 64 rows (256 KB)
    if (nBlocks > maxBlocks) nBlocks = maxBlocks;
    if (nBlocks < 1) nBlocks = 1;

    ce_softece_main<<<nBlocks, 256, 0, stream>>>(x, tgt, partials, nRows);
    ce_softece_final<<<1, 64, 0, stream>>>(partials, out, nBlocks, nRows);
}